// WeightedCoxNLL_22282290332222
// MI455X (gfx1250) — compile-verified
//
#include <hip/hip_runtime.h>
#include <hip/hip_bf16.h>
#include <stdint.h>

#define CENSORING 2.0f

typedef float v2f __attribute__((ext_vector_type(2)));
typedef float v8f __attribute__((ext_vector_type(8)));

// Wave-relative LDS byte address from a generic pointer to a __shared__ object:
// shared-aperture generic addresses carry the LDS offset in addr[31:0]
// (ISA 10.2).
__device__ __forceinline__ unsigned lds_addr32(const void* p) {
  return (unsigned)(uintptr_t)p;
}

// ---------------------------------------------------------------------------
// Stage 0: pack sortable keys.  t in (0.05, 1.0) is positive, so the IEEE-754
// bit pattern is order-isomorphic to the float value.  Low 32 bits carry the
// original index (== argsort payload).
// ---------------------------------------------------------------------------
__global__ void k_pack(const float* __restrict__ ytime,
                       unsigned long long* __restrict__ keys, int N) {
  int i = blockIdx.x * blockDim.x + threadIdx.x;
  if (i < N) {
    unsigned tb = __float_as_uint(ytime[i]);
    keys[i] = ((unsigned long long)tb << 32) | (unsigned)i;
  }
}

// ---------------------------------------------------------------------------
// Bitonic sort.  Global pass for stride j (cross-tile strides only).
// ---------------------------------------------------------------------------
__global__ void k_bitonic_global(unsigned long long* __restrict__ keys, int N,
                                 unsigned k, unsigned j) {
  unsigned tid = blockIdx.x * blockDim.x + threadIdx.x;
  if (tid >= (unsigned)(N >> 1)) return;
  unsigned l = ((tid & ~(j - 1u)) << 1) | (tid & (j - 1u));
  unsigned p = l | j;
  unsigned long long a = keys[l], b = keys[p];
  bool asc = ((l & k) == 0u);
  if ((a > b) == asc) { keys[l] = b; keys[p] = a; }
}

// LDS-resident passes: a 2048-element tile per block (16 KB of the 320 KB/WGP
// LDS).  Handles stages kFirst..kLast, each with all strides j <= 1024.
// Tile fill / writeback uses the CDNA5 async global<->LDS DMA path
// (ASYNCcnt-tracked, no VGPR staging).
__global__ __launch_bounds__(1024) void k_bitonic_shared(
    unsigned long long* __restrict__ keys, unsigned kFirst, unsigned kLast) {
  __shared__ unsigned long long lds[2048];
  const unsigned t = threadIdx.x;
  const unsigned base = blockIdx.x * 2048u;

  const unsigned l0 = lds_addr32(&lds[t]);
  const unsigned l1 = lds_addr32(&lds[t + 1024u]);
  const unsigned long long g0 = (unsigned long long)(uintptr_t)(keys + base + t);
  const unsigned long long g1 =
      (unsigned long long)(uintptr_t)(keys + base + t + 1024u);

  // global -> LDS, per-lane 8 bytes each, tracked by ASYNCcnt.
  asm volatile("global_load_async_to_lds_b64 %0, %1, off"
               :: "v"(l0), "v"(g0) : "memory");
  asm volatile("global_load_async_to_lds_b64 %0, %1, off"
               :: "v"(l1), "v"(g1) : "memory");
  asm volatile("s_wait_asynccnt 0x0" ::: "memory");
  __syncthreads();

  for (unsigned k = kFirst; k <= kLast; k <<= 1) {
    unsigned j0 = ((k >> 1) < 1024u) ? (k >> 1) : 1024u;
    for (unsigned j = j0; j > 0u; j >>= 1) {
      unsigned l = ((t & ~(j - 1u)) << 1) | (t & (j - 1u));
      unsigned p = l | j;
      unsigned long long a = lds[l], b = lds[p];
      bool asc = (((base + l) & k) == 0u);   // direction from *global* index
      if ((a > b) == asc) { lds[l] = b; lds[p] = a; }
      __syncthreads();
    }
  }

  // LDS -> global writeback, also async.
  asm volatile("global_store_async_from_lds_b64 %0, %1, off"
               :: "v"(g0), "v"(l0) : "memory");
  asm volatile("global_store_async_from_lds_b64 %0, %1, off"
               :: "v"(g1), "v"(l1) : "memory");
  asm volatile("s_wait_asynccnt 0x0" ::: "memory");
}

// ---------------------------------------------------------------------------
// Gather payloads in sorted order; compute s = t * exp(h).
// ---------------------------------------------------------------------------
__global__ void k_gather(const unsigned long long* __restrict__ keys,
                         const float* __restrict__ pred,
                         const unsigned char* __restrict__ ev,  // numpy bool: 1 byte
                         float* __restrict__ t_s, float* __restrict__ h_s,
                         float* __restrict__ e_s, float* __restrict__ s_s, int N) {
  int i = blockIdx.x * blockDim.x + threadIdx.x;
  if (i < N) {
    unsigned long long key = keys[i];
    float t = __uint_as_float((unsigned)(key >> 32));
    unsigned idx = (unsigned)(key & 0xFFFFFFFFu);
    float h = pred[idx];
    t_s[i] = t;
    h_s[i] = h;
    e_s[i] = ev[idx] ? 1.0f : 0.0f;
    s_s[i] = t * expf(h);
  }
}

// Zero-pad the 16-entry chunk-sum buffer so the WMMA scan kernel can load it
// unconditionally (keeps EXEC all-ones with zero divergence around the WMMAs).
__global__ void k_zero16(float* __restrict__ bsums) {
  unsigned t = threadIdx.x;
  if (t < 16u) bsums[t] = 0.0f;
}

// ---------------------------------------------------------------------------
// Suffix-scan level 1: per-chunk (1024 elements) totals, fixed-order LDS tree.
// ---------------------------------------------------------------------------
__global__ __launch_bounds__(1024) void k_blocksum(const float* __restrict__ s,
                                                   float* __restrict__ bsums) {
  __shared__ float red[1024];
  unsigned t = threadIdx.x;
  red[t] = s[blockIdx.x * 1024u + t];
  __syncthreads();
  for (unsigned off = 512u; off > 0u; off >>= 1) {
    if (t < off) red[t] += red[t + off];
    __syncthreads();
  }
  if (t == 0u) bsums[blockIdx.x] = red[0];
}

// ---------------------------------------------------------------------------
// Suffix-scan level 2 via WMMA: incl[m] = sum_{n>=m} bs[n] = (U * x)[m] with U
// the 16x16 upper-triangular ones matrix, K accumulated in 4 chunks of 4 using
// V_WMMA_F32_16X16X4_F32 chained through C.  Single wave32; straight-line code
// so EXEC is all ones at every WMMA.  offs[m] = incl[m] - bs[m].
//
// Layouts (ISA 7.12.2):
//   A 16x4 f32: lane L -> row M=L&15; VGPR0/1 hold K={0,1} (lanes 0-15) or
//               K={2,3} (lanes 16-31).
//   B 4x16 f32: column-constant here (B[k][n] = x[k]), same K-per-half split.
//   D: acc[r] = row r (lanes 0-15) / row r+8 (lanes 16-31); all columns equal.
// ---------------------------------------------------------------------------
__global__ __launch_bounds__(32) void k_suffix16_wmma(
    const float* __restrict__ bsums, float* __restrict__ offs) {
  const int lane = (int)threadIdx.x;      // 0..31
  const int m = lane & 15;                // A-matrix row owned by this lane
  const int khalf = (lane >> 4) << 1;     // +2 for lanes 16..31
  v8f acc = {};
#pragma unroll
  for (int c = 0; c < 4; ++c) {
    const int k0 = 4 * c + khalf;
    v2f A, B;
    A.x = (k0 >= m) ? 1.0f : 0.0f;        // U[m][k0]
    A.y = (k0 + 1 >= m) ? 1.0f : 0.0f;    // U[m][k0+1]
    B.x = bsums[k0];                      // B[k][*] = x[k] (zero-padded)
    B.y = bsums[k0 + 1];
    acc = __builtin_amdgcn_wmma_f32_16x16x4_f32(
        /*neg_a=*/false, A, /*neg_b=*/false, B,
        /*c_mod=*/(short)0, acc, /*reuse_a=*/false, /*reuse_b=*/false);
  }
  if (lane == 0) {
#pragma unroll
    for (int r = 0; r < 8; ++r) offs[r] = acc[r] - bsums[r];
  }
  if (lane == 16) {
#pragma unroll
    for (int r = 0; r < 8; ++r) offs[8 + r] = acc[r] - bsums[8 + r];
  }
}

// ---------------------------------------------------------------------------
// Suffix-scan level 3: per-chunk inclusive suffix scan (double-buffered
// Hillis-Steele in LDS) + later-chunk offset  ->  rev_cumsum.
// ---------------------------------------------------------------------------
__global__ __launch_bounds__(1024) void k_scan_block(
    const float* __restrict__ s, const float* __restrict__ offs,
    float* __restrict__ suff) {
  __shared__ float buf[2][1024];
  unsigned t = threadIdx.x;
  unsigned base = blockIdx.x * 1024u;
  buf[0][t] = s[base + t];
  __syncthreads();
  int src = 0;
  for (unsigned d = 1u; d < 1024u; d <<= 1) {
    float v = buf[src][t];
    if (t + d < 1024u) v += buf[src][t + d];
    buf[src ^ 1][t] = v;
    src ^= 1;
    __syncthreads();
  }
  suff[base + t] = buf[src][t] + offs[blockIdx.x];
}

// ---------------------------------------------------------------------------
// Per-element log-likelihood term + deterministic per-block partial sums.
// sum_exp[i] = (rev_cumsum[i] - s[i]) / t[i];  cens = (t < 2) & e;
// ll[i] = (h[i] - log(sum_exp[i])) * cens[i] for i < N-1;  n = sum(cens).
// ---------------------------------------------------------------------------
__global__ __launch_bounds__(1024) void k_term(
    const float* __restrict__ t_s, const float* __restrict__ h_s,
    const float* __restrict__ e_s, const float* __restrict__ s_s,
    const float* __restrict__ suff, float* __restrict__ pLL,
    float* __restrict__ pN, int N) {
  __shared__ float rll[1024];
  __shared__ float rn[1024];
  unsigned t = threadIdx.x;
  unsigned i = blockIdx.x * 1024u + t;
  float ll = 0.0f, cn = 0.0f;
  if ((int)i < N) {
    float tt = t_s[i];
    float cens = ((tt < CENSORING) && (e_s[i] != 0.0f)) ? 1.0f : 0.0f;
    cn = cens;
    if ((int)i < N - 1) {
      float sum_exp = (suff[i] - s_s[i]) / tt;
      ll = (h_s[i] - logf(sum_exp)) * cens;
    }
  }
  rll[t] = ll;
  rn[t] = cn;
  __syncthreads();
  for (unsigned off = 512u; off > 0u; off >>= 1) {
    if (t < off) { rll[t] += rll[t + off]; rn[t] += rn[t + off]; }
    __syncthreads();
  }
  if (t == 0u) { pLL[blockIdx.x] = rll[0]; pN[blockIdx.x] = rn[0]; }
}

__global__ void k_final(const float* __restrict__ pLL,
                        const float* __restrict__ pN, int P,
                        float* __restrict__ out) {
  if (threadIdx.x == 0 && blockIdx.x == 0) {
    float sll = 0.0f, sn = 0.0f;
    for (int i = 0; i < P; ++i) { sll += pLL[i]; sn += pN[i]; }
    out[0] = -sll / sn;
  }
}

// ---------------------------------------------------------------------------
// Launcher: inputs per reference setup order: pred(f32), ytime(f32),
// event_status(bool -> 1 byte).  Output: one f32 scalar.
// ---------------------------------------------------------------------------
extern "C" void kernel_launch(void* const* d_in, const int* in_sizes, int n_in,
                              void* d_out, int out_size, void* d_ws,
                              size_t ws_size, hipStream_t stream) {
  const float* pred = (const float*)d_in[0];
  const float* ytime = (const float*)d_in[1];
  const unsigned char* ev = (const unsigned char*)d_in[2];
  float* out = (float*)d_out;
  const int N = in_sizes[0];  // 16384 (power of two; bitonic assumes this)
  const int P = N / 1024;     // 16 scan chunks -> one 16-wide WMMA scan

  // Workspace layout (all 4/8-byte aligned; ~460 KB total for N=16384).
  char* ws = (char*)d_ws;
  unsigned long long* keys = (unsigned long long*)ws;
  float* t_s = (float*)(ws + (size_t)N * 8u);
  float* h_s = t_s + N;
  float* e_s = h_s + N;
  float* s_s = e_s + N;
  float* suff = s_s + N;
  float* bsums = suff + N;   // 16 entries, zero-padded
  float* offs = bsums + 16;
  float* pLL = offs + 16;
  float* pN = pLL + 16;

  dim3 b256(256);
  dim3 gN((N + 255) / 256);
  dim3 gHalf((N / 2 + 255) / 256);

  k_pack<<<gN, b256, 0, stream>>>(ytime, keys, N);

  // Bitonic sort: stages k=2..2048 fully LDS-resident (async DMA tile
  // fill/writeback), then for each larger stage the cross-tile strides
  // globally and the tail (j<=1024) in LDS.
  k_bitonic_shared<<<N / 2048, 1024, 0, stream>>>(keys, 2u, 2048u);
  for (unsigned k = 4096u; k <= (unsigned)N; k <<= 1) {
    for (unsigned j = k >> 1; j >= 2048u; j >>= 1)
      k_bitonic_global<<<gHalf, b256, 0, stream>>>(keys, N, k, j);
    k_bitonic_shared<<<N / 2048, 1024, 0, stream>>>(keys, k, k);
  }

  k_gather<<<gN, b256, 0, stream>>>(keys, pred, ev, t_s, h_s, e_s, s_s, N);

  // Three-level suffix scan (level 2 is the WMMA triangular-matrix scan).
  k_zero16<<<1, 32, 0, stream>>>(bsums);
  k_blocksum<<<P, 1024, 0, stream>>>(s_s, bsums);
  k_suffix16_wmma<<<1, 32, 0, stream>>>(bsums, offs);
  k_scan_block<<<P, 1024, 0, stream>>>(s_s, offs, suff);

  k_term<<<P, 1024, 0, stream>>>(t_s, h_s, e_s, s_s, suff, pLL, pN, N);
  k_final<<<1, 32, 0, stream>>>(pLL, pN, P, out);
}